// SSIM_87290915323963
// MI455X (gfx1250) — compile-verified
//
#include <hip/hip_runtime.h>
#include <math.h>

// SSIM via banded-matmul Gaussian blur on CDNA5 WMMA (fp32 16x16x4).
//
// blur(X) = G * X * G^T, G[i,j] = w[j-i+40] (512x512, band 81, symmetric).
// Banded K: a 16-wide output tile needs only K = 16 + 2*40 = 96 -> 24 WMMA
// k-steps of V_WMMA_F32_16X16X4_F32 per tile per channel. fp32 WMMA is chosen
// over f16/bf16 because sigma = E[x^2] - mu^2 cancels; low-precision A/B would
// corrupt the SSIM map, and at ~8 GFLOP of banded matmul the matrix pipe is
// still far ahead of the ~13.6 GFLOP scalar-VALU direct convolution.
//
// Workspace: 5 planes x 16 images x 512 x 512 fp32 = 83,886,080 bytes.

typedef __attribute__((ext_vector_type(2))) float v2f;
typedef __attribute__((ext_vector_type(8))) float v8f;

#define IMG_H 512
#define IMG_W 512
#define NIMG  16
#define WIN   81
#define PAD   40          // WIN/2
#define BAND  96          // 16 + 2*PAD
#define KSTEPS 24         // BAND/4
#define XS_STRIDE 80      // LDS row stride for pass1 x tiles (bank spread)
#define TL_STRIDE 145     // LDS row stride for pass2 T tiles (conflict-free)

__device__ __forceinline__ float cleanf(float v) {
    v = (v != v) ? 0.0f : v;                 // nan -> 0
    return fminf(fmaxf(v, 0.0f), 1.0f);      // clip handles +-inf
}

// ---------------------------------------------------------------------------
// Pass 1: vertical blur T_c = G * X_c for the 5 channels, fused preprocessing.
// Block 128 = 4 waves. Block tile: 16 output rows x 64 cols; wave: 16 cols.
// ---------------------------------------------------------------------------
__global__ __launch_bounds__(128) void ssim_pass1(
    const float* __restrict__ img1, const float* __restrict__ img2,
    const float* __restrict__ w1d, float* __restrict__ T)
{
    __shared__ float wpad[112];                 // w zero-padded: wpad[16+k]=w[k]
    __shared__ float xs1[BAND * XS_STRIDE];
    __shared__ float xs2[BAND * XS_STRIDE];

    const int tid = threadIdx.x;
    const int cg0 = blockIdx.x * 64;            // column group base
    const int m0  = blockIdx.y * 16;            // output row tile base
    const int img = blockIdx.z;

    if (tid < 112) wpad[tid] = (tid >= 16 && tid < 16 + WIN) ? w1d[tid - 16] : 0.0f;

    const float* p1 = img1 + (size_t)img * IMG_H * IMG_W;
    const float* p2 = img2 + (size_t)img * IMG_H * IMG_W;
    // Stage cleaned 96x64 input band (zero-padded rows outside the image).
    for (int i = tid; i < BAND * 64; i += 128) {      // 48 exact iters/thread
        int r = i >> 6, c = i & 63;
        int gr = m0 - PAD + r;
        float a = 0.0f, b = 0.0f;
        if (gr >= 0 && gr < IMG_H) {
            int off = gr * IMG_W + cg0 + c;
            a = cleanf(p1[off]);
            b = cleanf(p2[off]);
        }
        xs1[r * XS_STRIDE + c] = a;
        xs2[r * XS_STRIDE + c] = b;
    }
    __syncthreads();

    const int lane = tid & 31;
    const int wv   = tid >> 5;
    const int j    = lane & 15;     // A row / B col within tile
    const int hi   = lane >> 4;     // K half-select (A & B share kk = v + 2*hi)

    v8f acc0 = {}, acc1 = {}, acc2 = {}, acc3 = {}, acc4 = {};

    for (int step = 0; step < KSTEPS; ++step) {
        const int kb = step << 2;
        v2f a, b1, b2, b11, b22, b12;
#pragma unroll
        for (int v = 0; v < 2; ++v) {
            const int kk = v + 2 * hi;
            // A = G[m0+j][kglob+kk] = w[(kb+kk) - j]; index in [1,111].
            a[v] = wpad[16 + kb + kk - j];
            const int xi = (kb + kk) * XS_STRIDE + wv * 16 + j;
            const float x1 = xs1[xi], x2 = xs2[xi];
            b1[v] = x1;  b2[v] = x2;
            b11[v] = x1 * x1;  b22[v] = x2 * x2;  b12[v] = x1 * x2;
        }
        acc0 = __builtin_amdgcn_wmma_f32_16x16x4_f32(false, a, false, b1,  (short)0, acc0, false, false);
        acc1 = __builtin_amdgcn_wmma_f32_16x16x4_f32(false, a, false, b2,  (short)0, acc1, false, false);
        acc2 = __builtin_amdgcn_wmma_f32_16x16x4_f32(false, a, false, b11, (short)0, acc2, false, false);
        acc3 = __builtin_amdgcn_wmma_f32_16x16x4_f32(false, a, false, b22, (short)0, acc3, false, false);
        acc4 = __builtin_amdgcn_wmma_f32_16x16x4_f32(false, a, false, b12, (short)0, acc4, false, false);
    }

    // C/D layout: lane L, VGPR v -> row m = v + 8*(L>=16), col = L%16.
    const int col = cg0 + wv * 16 + j;
    const size_t plane = (size_t)IMG_H * IMG_W;
#pragma unroll
    for (int v = 0; v < 8; ++v) {
        const size_t rowoff = (size_t)(m0 + v + 8 * hi) * IMG_W + col;
        T[(0 * NIMG + img) * plane + rowoff] = acc0[v];
        T[(1 * NIMG + img) * plane + rowoff] = acc1[v];
        T[(2 * NIMG + img) * plane + rowoff] = acc2[v];
        T[(3 * NIMG + img) * plane + rowoff] = acc3[v];
        T[(4 * NIMG + img) * plane + rowoff] = acc4[v];
    }
}

// ---------------------------------------------------------------------------
// Pass 2: horizontal blur Y_c = T_c * G, fused SSIM combine on accumulators.
// ---------------------------------------------------------------------------
__global__ __launch_bounds__(128) void ssim_pass2(
    const float* __restrict__ T, const float* __restrict__ w1d,
    float* __restrict__ out)
{
    __shared__ float wpad[112];
    __shared__ float tl[5 * 16 * TL_STRIDE];    // [channel][row 16][band col 144(+1)]

    const int tid = threadIdx.x;
    const int cg0 = blockIdx.x * 64;
    const int m0  = blockIdx.y * 16;
    const int img = blockIdx.z;

    if (tid < 112) wpad[tid] = (tid >= 16 && tid < 16 + WIN) ? w1d[tid - 16] : 0.0f;

    const size_t plane = (size_t)IMG_H * IMG_W;
    // Stage 5 channels x 16 rows x 144 band cols of T (zero-padded cols).
    for (int i = tid; i < 5 * 16 * 144; i += 128) {   // 90 exact iters/thread
        int c   = i / 2304;
        int rem = i - c * 2304;
        int r   = rem / 144;
        int cc  = rem - r * 144;
        int gc  = cg0 - PAD + cc;
        float v = 0.0f;
        if (gc >= 0 && gc < IMG_W)
            v = T[((size_t)(c * NIMG + img)) * plane + (size_t)(m0 + r) * IMG_W + gc];
        tl[c * (16 * TL_STRIDE) + r * TL_STRIDE + cc] = v;
    }
    __syncthreads();

    const int lane = tid & 31;
    const int wv   = tid >> 5;
    const int j    = lane & 15;
    const int hi   = lane >> 4;

    v8f acc0 = {}, acc1 = {}, acc2 = {}, acc3 = {}, acc4 = {};

    for (int step = 0; step < KSTEPS; ++step) {
        const int kb = step << 2;
        v2f b, a0, a1, a2, a3, a4;
#pragma unroll
        for (int v = 0; v < 2; ++v) {
            const int kk = v + 2 * hi;
            // B = G[kglob+kk][n0+j] = w[j - kb - kk + 80]; index in [1,111].
            b[v] = wpad[96 + j - kb - kk];
            const int ci = j * TL_STRIDE + wv * 16 + kb + kk;   // A = T[m0+j][band]
            a0[v] = tl[ci];
            a1[v] = tl[1 * (16 * TL_STRIDE) + ci];
            a2[v] = tl[2 * (16 * TL_STRIDE) + ci];
            a3[v] = tl[3 * (16 * TL_STRIDE) + ci];
            a4[v] = tl[4 * (16 * TL_STRIDE) + ci];
        }
        acc0 = __builtin_amdgcn_wmma_f32_16x16x4_f32(false, a0, false, b, (short)0, acc0, false, false);
        acc1 = __builtin_amdgcn_wmma_f32_16x16x4_f32(false, a1, false, b, (short)0, acc1, false, false);
        acc2 = __builtin_amdgcn_wmma_f32_16x16x4_f32(false, a2, false, b, (short)0, acc2, false, false);
        acc3 = __builtin_amdgcn_wmma_f32_16x16x4_f32(false, a3, false, b, (short)0, acc3, false, false);
        acc4 = __builtin_amdgcn_wmma_f32_16x16x4_f32(false, a4, false, b, (short)0, acc4, false, false);
    }

    // All 5 accumulators share the same C/D lane layout -> SSIM combine is
    // lane-local on the accumulator components.
    const float C1v = 1e-4f, C2v = 9e-4f;
    const int col = cg0 + wv * 16 + j;
#pragma unroll
    for (int v = 0; v < 8; ++v) {
        const float mu1 = acc0[v], mu2 = acc1[v];
        const float e11 = acc2[v], e22 = acc3[v], e12 = acc4[v];
        const float mu1sq = mu1 * mu1, mu2sq = mu2 * mu2, mu12 = mu1 * mu2;
        const float s11 = fmaxf(e11 - mu1sq, 0.0f);
        const float s22 = fmaxf(e22 - mu2sq, 0.0f);
        const float s12 = e12 - mu12;
        const float num = (2.0f * mu12 + C1v) * (2.0f * s12 + C2v);
        const float den = (mu1sq + mu2sq + C1v) * (s11 + s22 + C2v);
        float r = num / (den + 1e-8f);
        r = (r != r) ? 0.0f : r;
        r = fminf(fmaxf(r, -1.0f), 1.0f);
        out[(size_t)(img * IMG_H + m0 + v + 8 * hi) * IMG_W + col] = r;
    }
}

extern "C" void kernel_launch(void* const* d_in, const int* in_sizes, int n_in,
                              void* d_out, int out_size, void* d_ws, size_t ws_size,
                              hipStream_t stream) {
    const float* img1 = (const float*)d_in[0];   // 16x1x512x512 fp32
    const float* img2 = (const float*)d_in[1];   // 16x1x512x512 fp32
    const float* w1d  = (const float*)d_in[2];   // 81 fp32
    float* T   = (float*)d_ws;                   // 5 planes x 16 imgs x 512 x 512
    float* out = (float*)d_out;

    dim3 grid(IMG_W / 64, IMG_H / 16, NIMG);     // 8 x 32 x 16
    dim3 block(128);
    hipLaunchKernelGGL(ssim_pass1, grid, block, 0, stream, img1, img2, w1d, T);
    hipLaunchKernelGGL(ssim_pass2, grid, block, 0, stream, T, w1d, out);
}